// Model_1331439862418
// MI455X (gfx1250) — compile-verified
//
#include <hip/hip_runtime.h>
#include <hip/hip_bf16.h>
#include <cstddef>

typedef __attribute__((ext_vector_type(2))) float v2f;
typedef __attribute__((ext_vector_type(8))) float v8f;

#define B_TOTAL 512
#define POOL_K 7
#define POOL_S 5

__device__ __forceinline__ float fast_tanh(float x) {
#if __has_builtin(__builtin_amdgcn_tanhf)
    return __builtin_amdgcn_tanhf(x);          // native v_tanh_f32 (1 TRANS op)
#elif __has_builtin(__builtin_amdgcn_exp2f) && __has_builtin(__builtin_amdgcn_rcpf)
    float e = __builtin_amdgcn_exp2f(2.8853900817779268f * x);
    return 1.0f - 2.0f * __builtin_amdgcn_rcpf(e + 1.0f);
#else
    return tanhf(x);
#endif
}

// Fused RNN layer: h = tanh(in[:,t,:] @ W_ih^T + b_ih + b_hh + h @ W_hh^T),
// ReLU + avgpool(K=7,S=5) folded in; writes pooled output [B][Lout][H].
// One workgroup per 16-row batch tile; NW = H/16 waves; weights in VGPR
// B-fragments; h double-buffered in LDS (WMMA D->A layout conversion) so a
// single minimal sync per step suffices: s_wait_dscnt + split barrier for
// NW>1, and nothing at all for single-wave blocks (DS pipe is in-order per
// wave). No loadcnt/storecnt fences -> input prefetch and pooled-output
// stores stay in flight across steps.
template<int I, int H, int NW>
__global__ __launch_bounds__(32 * NW)
void rnn_scan_kernel(const float* __restrict__ in,    // [B][T][I]
                     const float* __restrict__ w_ih,  // [H][I]
                     const float* __restrict__ w_hh,  // [H][H]
                     const float* __restrict__ b_ih,  // [H]
                     const float* __restrict__ b_hh,  // [H]
                     float* __restrict__ out,         // [B][Lout][H]
                     int T, int Lout)
{
    constexpr int KI = (I + 3) / 4;   // K-fragments for input projection
    constexpr int KH = H / 4;         // K-fragments for recurrence
    constexpr int KG = KI + KH;       // total WMMA ops per step

    __shared__ float hbuf[2 * 16 * H];  // ping-pong h state: [buf][m][H]

    const int lane = threadIdx.x & 31;
    const int wave = threadIdx.x >> 5;
    const int nb   = wave * 16;           // output-column base for this wave
    const int b0   = blockIdx.x * 16;     // batch-row base for this block
    const int nl   = lane & 15;           // N index (B/C/D) == M index (A)
    const int half = lane >> 4;           // 0 or 1 (selects K pair / M+8)

    // ---- Preload WMMA B-fragments of W_ih^T and W_hh^T (columns nb..nb+15)
    // B layout (4x16 f32): lane%16 = N, K = vgpr + 2*(lane/16)
    v2f bi[KI];
    float imask[KI][2];               // per-lane 0/1 mask for boundary comps
#pragma unroll
    for (int f = 0; f < KI; ++f) {
#pragma unroll
        for (int v = 0; v < 2; ++v) {
            int k = f * 4 + v + 2 * half;
            bi[f][v] = (k < I) ? w_ih[(size_t)(nb + nl) * I + k] : 0.0f;
            imask[f][v] = (k < I) ? 1.0f : 0.0f;
        }
    }
    v2f bh[KH];
#pragma unroll
    for (int f = 0; f < KH; ++f) {
#pragma unroll
        for (int v = 0; v < 2; ++v) {
            int k = f * 4 + v + 2 * half;
            bh[f][v] = w_hh[(size_t)(nb + nl) * H + k];
        }
    }

    // loop-invariant C operands for the two accumulator chains
    v8f cbias, zero8;
    {
        const float bias = b_ih[nb + nl] + b_hh[nb + nl];
#pragma unroll
        for (int r = 0; r < 8; ++r) { cbias[r] = bias; zero8[r] = 0.0f; }
    }

    // ---- raw input pipeline: load WITHOUT masking (mask applied at use)
    float rawin[KI][2], rawin_nxt[KI][2];
    auto load_raw = [&](float (*a)[2], int t) {
        const float* row = in + ((size_t)(b0 + nl) * T + t) * I;
#pragma unroll
        for (int f = 0; f < KI; ++f) {
#pragma unroll
            for (int v = 0; v < 2; ++v) {
                if (f * 4 + v < I) {              // folds at compile time
                    int k = f * 4 + v + 2 * half;
                    int kc = (k < I) ? k : 0;     // clamped, always valid
                    a[f][v] = row[kc];
                }
            }
        }
    };
    auto make_ain = [&](int f) -> v2f {           // mask at use site
        v2f a;
#pragma unroll
        for (int v = 0; v < 2; ++v) {
            if (f * 4 + v >= I)          a[v] = 0.0f;
            else if (f * 4 + v + 2 < I)  a[v] = rawin[f][v];
            else                         a[v] = rawin[f][v] * imask[f][v];
        }
        return a;
    };

    // ---- h_{-1} = 0 lives in buffer 1 (step t reads buf (t+1)&1)
    for (int i = threadIdx.x; i < 2 * 16 * H; i += blockDim.x) hbuf[i] = 0.0f;
    __syncthreads();

    load_raw(rawin, 0);

    // ---- one recurrence step: h_t from h_{t-1}; returns relu(h_t)
    auto step = [&](int t, bool pf) -> v8f {
        if (pf) load_raw(rawin_nxt, t + 1);

        const float* hr = &hbuf[((t + 1) & 1) * 16 * H];  // h_{t-1}
        float*       hw = &hbuf[(t & 1) * 16 * H];        // h_t

        auto getA = [&](int g) -> v2f {
            if (g < KI) return make_ain(g);
            int f = g - KI;
            v2f a;
            a[0] = hr[nl * H + f * 4 + 2 * half];
            a[1] = hr[nl * H + f * 4 + 1 + 2 * half];
            return a;
        };
        auto getB = [&](int g) -> v2f { return (g < KI) ? bi[g] : bh[g - KI]; };

        // two interleaved accumulator chains (halves dependent-WMMA depth)
        v8f c0 = __builtin_amdgcn_wmma_f32_16x16x4_f32(
                     false, getA(0), false, getB(0), (short)0, cbias, false, false);
        v8f c1 = __builtin_amdgcn_wmma_f32_16x16x4_f32(
                     false, getA(1), false, getB(1), (short)0, zero8, false, false);
#pragma unroll
        for (int g = 2; g < KG; ++g) {
            if (g & 1)
                c1 = __builtin_amdgcn_wmma_f32_16x16x4_f32(
                         false, getA(g), false, getB(g), (short)0, c1, false, false);
            else
                c0 = __builtin_amdgcn_wmma_f32_16x16x4_f32(
                         false, getA(g), false, getB(g), (short)0, c0, false, false);
        }

        // merge chains, tanh + relu
        v8f hnew, rel;
#pragma unroll
        for (int r = 0; r < 8; ++r) {
            float th = fast_tanh(c0[r] + c1[r]);
            hnew[r] = th;
            rel[r]  = th > 0.0f ? th : 0.0f;
        }
        // publish h_t (D layout: M = r + 8*half, N = lane%16)
#pragma unroll
        for (int r = 0; r < 8; ++r)
            hw[(r + 8 * half) * H + nb + nl] = hnew[r];

        // minimal cross-step sync: LDS-only. Reads (buf R) and writes (buf W)
        // are disjoint this step; next step swaps roles, so one sync point
        // after the stores is sufficient.
        if constexpr (NW > 1) {
            asm volatile(
                "s_wait_dscnt 0x0\n\t"
                "s_barrier_signal -1\n\t"
                "s_barrier_wait -1" ::: "memory");
        } else {
            // single wave: DS pipe is in-order; only stop compiler reordering
            asm volatile("" ::: "memory");
        }

        // rotate raw input pipeline
#pragma unroll
        for (int f = 0; f < KI; ++f) {
#pragma unroll
            for (int v = 0; v < 2; ++v)
                if (f * 4 + v < I) rawin[f][v] = rawin_nxt[f][v];
        }
        return rel;
    };

    const float inv_k = 1.0f / (float)POOL_K;
    const size_t orow = (size_t)Lout * H;   // out row stride for one batch

    v8f accY, accO;                         // young / old window accumulators
#pragma unroll
    for (int r = 0; r < 8; ++r) { accY[r] = 0.0f; accO[r] = 0.0f; }

    // ---- main chunked loop: chunk q = time steps [5q, 5q+5)
    // (t+1 <= 5*Lout < T inside chunks, so prefetch is unconditional)
    for (int q = 0; q < Lout; ++q) {
        const int tb = q * POOL_S;
        v8f r0 = step(tb,     true);
        v8f r1 = step(tb + 1, true);
        if (q > 0) {            // window q-1: ages 5,6 -> close & store
#pragma unroll
            for (int r = 0; r < 8; ++r) accO[r] += r0[r] + r1[r];
#pragma unroll
            for (int r = 0; r < 8; ++r)
                out[(size_t)(b0 + r + 8 * half) * orow + (size_t)(q - 1) * H + nb + nl]
                    = accO[r] * inv_k;
        }
#pragma unroll
        for (int r = 0; r < 8; ++r) accY[r] = r0[r] + r1[r];   // window q: ages 0,1
        v8f r2 = step(tb + 2, true);
        v8f r3 = step(tb + 3, true);
        v8f r4 = step(tb + 4, true);
#pragma unroll
        for (int r = 0; r < 8; ++r) accY[r] += r2[r] + r3[r] + r4[r];
#pragma unroll
        for (int r = 0; r < 8; ++r) accO[r] = accY[r];         // rotate roles
    }

    // ---- tail steps: finish last window (ages 5.. of window Lout-1)
    const int tail = T - POOL_S * Lout;     // == 2 for all layers here
    for (int j = 0; j < tail; ++j) {
        const int tt = POOL_S * Lout + j;
        v8f rr = step(tt, tt + 1 < T);
        if (j < POOL_K - POOL_S) {          // age 5+j < K
#pragma unroll
            for (int r = 0; r < 8; ++r) accO[r] += rr[r];
            if (j == POOL_K - POOL_S - 1) { // age == K-1: close last window
#pragma unroll
                for (int r = 0; r < 8; ++r)
                    out[(size_t)(b0 + r + 8 * half) * orow + (size_t)(Lout - 1) * H + nb + nl]
                        = accO[r] * inv_k;
            }
        }
    }
}

// Final FC: out[b][o] = fc_b[o] + sum_j flat[b][j] * fc_w[o][j]
__global__ void fc_kernel(const float* __restrict__ flat,  // [512][640]
                          const float* __restrict__ w,     // [10][640]
                          const float* __restrict__ bias,  // [10]
                          float* __restrict__ out)         // [512][10]
{
    int idx = blockIdx.x * blockDim.x + threadIdx.x;
    if (idx >= B_TOTAL * 10) return;
    int b = idx / 10, o = idx % 10;
    float s = bias[o];
    const float* fr = flat + (size_t)b * 640;
    const float* wr = w + (size_t)o * 640;
#pragma unroll 8
    for (int j = 0; j < 640; ++j) s += fr[j] * wr[j];
    out[idx] = s;
}

extern "C" void kernel_launch(void* const* d_in, const int* in_sizes, int n_in,
                              void* d_out, int out_size, void* d_ws, size_t ws_size,
                              hipStream_t stream)
{
    const float* x = (const float*)d_in[0];
    const float* w_ih1 = (const float*)d_in[1];
    const float* w_hh1 = (const float*)d_in[2];
    const float* b_ih1 = (const float*)d_in[3];
    const float* b_hh1 = (const float*)d_in[4];
    const float* w_ih2 = (const float*)d_in[5];
    const float* w_hh2 = (const float*)d_in[6];
    const float* b_ih2 = (const float*)d_in[7];
    const float* b_hh2 = (const float*)d_in[8];
    const float* w_ih3 = (const float*)d_in[9];
    const float* w_hh3 = (const float*)d_in[10];
    const float* b_ih3 = (const float*)d_in[11];
    const float* b_hh3 = (const float*)d_in[12];
    const float* w_ih4 = (const float*)d_in[13];
    const float* w_hh4 = (const float*)d_in[14];
    const float* b_ih4 = (const float*)d_in[15];
    const float* b_hh4 = (const float*)d_in[16];
    const float* fc_w  = (const float*)d_in[17];
    const float* fc_b  = (const float*)d_in[18];

    // Time lengths through the stack (each satisfies T = 5*Lout + 2)
    const int T1 = 3437, L1 = (T1 - POOL_K) / POOL_S + 1;   // 687
    const int T2 = L1,   L2 = (T2 - POOL_K) / POOL_S + 1;   // 137
    const int T3 = L2,   L3 = (T3 - POOL_K) / POOL_S + 1;   // 27
    const int T4 = L3,   L4 = (T4 - POOL_K) / POOL_S + 1;   // 5

    // Workspace: bufA holds layer-1 (largest) & layer-3 outputs,
    // bufB holds layer-2 & layer-4 outputs (ping-pong). ~32 MB total.
    float* bufA = (float*)d_ws;                              // >= 512*687*16
    float* bufB = bufA + (size_t)B_TOTAL * L1 * 16;          // >= 512*137*32

    const int NBLK = B_TOTAL / 16;   // 32 batch tiles

    rnn_scan_kernel<1, 16, 1><<<NBLK, 32, 0, stream>>>(
        x, w_ih1, w_hh1, b_ih1, b_hh1, bufA, T1, L1);
    rnn_scan_kernel<16, 32, 2><<<NBLK, 64, 0, stream>>>(
        bufA, w_ih2, w_hh2, b_ih2, b_hh2, bufB, T2, L2);
    rnn_scan_kernel<32, 64, 4><<<NBLK, 128, 0, stream>>>(
        bufB, w_ih3, w_hh3, b_ih3, b_hh3, bufA, T3, L3);
    rnn_scan_kernel<64, 128, 8><<<NBLK, 256, 0, stream>>>(
        bufA, w_ih4, w_hh4, b_ih4, b_hh4, bufB, T4, L4);
    // bufB now holds [512][5][128] == flattened [512][640]
    fc_kernel<<<(B_TOTAL * 10 + 255) / 256, 256, 0, stream>>>(
        bufB, fc_w, fc_b, (float*)d_out);
}